// EdgeSAGERNN_14302241096332
// MI455X (gfx1250) — compile-verified
//
#include <hip/hip_runtime.h>
#include <hip/hip_bf16.h>
#include <math.h>

#define DIM 128          // D == H == 128
#define LDS_STRIDE 132   // padded row stride (floats) to avoid bank conflicts

typedef float v2f __attribute__((ext_vector_type(2)));
typedef float v8f __attribute__((ext_vector_type(8)));

// ---------------------------------------------------------------------------
// Zero a float buffer
// ---------------------------------------------------------------------------
__global__ void zero_kernel(float* __restrict__ p, long long n) {
    long long i = (long long)blockIdx.x * blockDim.x + threadIdx.x;
    if (i < n) p[i] = 0.0f;
}

// ---------------------------------------------------------------------------
// Edge scatter: one wave32 per edge; 4 floats per lane (128 = 32*4).
// agg[dst] += X[src]; optionally cnt[dst] += 1 (lane 0).
// unsafeAtomicAdd -> global_atomic_add_f32 (no-return HW float atomic).
// ---------------------------------------------------------------------------
__global__ void scatter_kernel(const float* __restrict__ X,
                               const int* __restrict__ src,
                               const int* __restrict__ dst,
                               float* __restrict__ agg,
                               float* __restrict__ cnt,
                               long long E, int do_count) {
    long long t = (long long)blockIdx.x * blockDim.x + threadIdx.x;
    long long e = t >> 5;
    int lane = (int)(t & 31);
    if (e >= E) return;
    int s = src[e];
    int d = dst[e];
    const float4 v = *(const float4*)(X + (size_t)s * DIM + lane * 4);
    float* p = agg + (size_t)d * DIM + lane * 4;
    unsafeAtomicAdd(p + 0, v.x);
    unsafeAtomicAdd(p + 1, v.y);
    unsafeAtomicAdd(p + 2, v.z);
    unsafeAtomicAdd(p + 3, v.w);
    if (do_count && lane == 0) unsafeAtomicAdd(cnt + d, 1.0f);
}

// ---------------------------------------------------------------------------
// Fused WMMA GEMM:
//   Y[r][c] = act( A1s[r][:] @ W1[c][:]  (+ A2[r][:] @ W2[c][:])  + bias[c] (+ bias2[c]) )
// where A1s = A1 row-scaled by 1/max(cnt,1) if cnt != nullptr.
// W matrices are [128 x 128] row-major and used transposed (Y += A @ W^T).
// Block = 256 threads = 8 waves; block owns 16 rows, wave w owns cols [16w,16w+16).
// K accumulated with V_WMMA_F32_16X16X4_F32 (f32-precise, matches f32 reference).
// act: 0 = none, 1 = relu, 2 = tanh
// ---------------------------------------------------------------------------
__global__ void gemm_wmma_kernel(const float* __restrict__ A1,
                                 const float* __restrict__ cnt,
                                 const float* __restrict__ W1,
                                 const float* __restrict__ A2,
                                 const float* __restrict__ W2,
                                 const float* __restrict__ bias,
                                 const float* __restrict__ bias2,
                                 float* __restrict__ Y,
                                 int nrows, int act) {
    __shared__ float ldsA[16 * LDS_STRIDE];
    __shared__ float ldsB[16 * LDS_STRIDE];

    const int tid = threadIdx.x;
    const int rowBase = blockIdx.x * 16;

    // ---- Stage 16x128 row block(s) into LDS (2 float4s per thread per mat) ----
    for (int i = tid; i < 16 * 32; i += 256) {        // 512 float4 elements
        int r = i >> 5;                                // row within tile
        int c = (i & 31) << 2;                         // float column
        int row = rowBase + r;
        int rowc = row < nrows ? row : (nrows - 1);    // clamp OOB reads
        float sc = 1.0f;
        if (cnt) sc = 1.0f / fmaxf(cnt[rowc], 1.0f);
        float4 v = *(const float4*)(A1 + (size_t)rowc * DIM + c);
        float* pa = ldsA + r * LDS_STRIDE + c;
        pa[0] = v.x * sc; pa[1] = v.y * sc; pa[2] = v.z * sc; pa[3] = v.w * sc;
        if (A2) {
            float4 u = *(const float4*)(A2 + (size_t)rowc * DIM + c);
            *(float4*)(ldsB + r * LDS_STRIDE + c) = u;
        }
    }
    __syncthreads();

    const int wave = tid >> 5;
    const int lane = tid & 31;
    const int r = lane & 15;        // row (A frag) / col (B,C frags) within tile
    const int half = lane >> 4;     // K-pair selector for A/B frags; row-bank for C/D
    const int colBase = wave * 16;
    const int col = colBase + r;

    // Bias broadcast: C/D element (v, lane) lives at row v+8*half, col (lane&15)
    float bv = 0.0f;
    if (bias)  bv += bias[col];
    if (bias2) bv += bias2[col];
    v8f acc;
#pragma unroll
    for (int v = 0; v < 8; ++v) acc[v] = bv;

    // ---- K loop over W1 (A = scaled mean rows) ----
    const float* w1row = W1 + (size_t)col * DIM;
#pragma unroll
    for (int k = 0; k < DIM; k += 4) {
        v2f a = *(const v2f*)(ldsA + r * LDS_STRIDE + k + 2 * half);
        v2f b = *(const v2f*)(w1row + k + 2 * half);   // B[k][col] = W1[col][k]
        acc = __builtin_amdgcn_wmma_f32_16x16x4_f32(
            /*neg_a=*/false, a, /*neg_b=*/false, b,
            /*c_mod=*/(short)0, acc, /*reuse_a=*/false, /*reuse_b=*/false);
    }

    // ---- Optional second product (root/self term) ----
    if (A2) {
        const float* w2row = W2 + (size_t)col * DIM;
#pragma unroll
        for (int k = 0; k < DIM; k += 4) {
            v2f a = *(const v2f*)(ldsB + r * LDS_STRIDE + k + 2 * half);
            v2f b = *(const v2f*)(w2row + k + 2 * half);
            acc = __builtin_amdgcn_wmma_f32_16x16x4_f32(
                false, a, false, b, (short)0, acc, false, false);
        }
    }

    // ---- Activation + store: VGPR v holds row (v + 8*half), col (lane&15) ----
#pragma unroll
    for (int v = 0; v < 8; ++v) {
        int row = rowBase + v + 8 * half;
        if (row < nrows) {
            float y = acc[v];
            if (act == 1)      y = fmaxf(y, 0.0f);
            else if (act == 2) y = tanhf(y);
            Y[(size_t)row * DIM + col] = y;
        }
    }
}

// ---------------------------------------------------------------------------
// Decode: one wave32 per query pair; 4 floats/lane; wave-shuffle reduction.
// sigmoid( dot(a,b) / (max(|a|,eps)*max(|b|,eps)) )
// ---------------------------------------------------------------------------
__global__ void decode_kernel(const float* __restrict__ out,
                              const int* __restrict__ dec,
                              long long Q,
                              float* __restrict__ res) {
    long long t = (long long)blockIdx.x * blockDim.x + threadIdx.x;
    long long q = t >> 5;
    int lane = (int)(t & 31);
    if (q >= Q) return;
    int ia = dec[q];
    int ib = dec[Q + q];
    const float4 a = *(const float4*)(out + (size_t)ia * DIM + lane * 4);
    const float4 b = *(const float4*)(out + (size_t)ib * DIM + lane * 4);
    float dot = a.x * b.x + a.y * b.y + a.z * b.z + a.w * b.w;
    float na  = a.x * a.x + a.y * a.y + a.z * a.z + a.w * a.w;
    float nb  = b.x * b.x + b.y * b.y + b.z * b.z + b.w * b.w;
#pragma unroll
    for (int off = 16; off > 0; off >>= 1) {
        dot += __shfl_xor(dot, off, 32);
        na  += __shfl_xor(na,  off, 32);
        nb  += __shfl_xor(nb,  off, 32);
    }
    if (lane == 0) {
        float da = fmaxf(sqrtf(na), 1e-6f);
        float db = fmaxf(sqrtf(nb), 1e-6f);
        float c = dot / (da * db);
        res[q] = 1.0f / (1.0f + expf(-c));
    }
}

// ---------------------------------------------------------------------------
// Launch: zero -> scatter(x)+count -> GEMM1(relu) -> zero -> scatter(h)
//         -> GEMM2 -> GEMM3(tanh, bi+bh) -> GEMM4 -> decode
// Workspace layout (floats): agg[N*128] | cnt[N] | h[N*128] | h2[N*128]
// hid reuses h; out reuses agg.
// ---------------------------------------------------------------------------
extern "C" void kernel_launch(void* const* d_in, const int* in_sizes, int n_in,
                              void* d_out, int out_size, void* d_ws, size_t ws_size,
                              hipStream_t stream) {
    const float* x   = (const float*)d_in[0];
    const int*   ei  = (const int*)d_in[1];
    const int*   dec = (const int*)d_in[2];
    const float* W1l = (const float*)d_in[3];
    const float* b1l = (const float*)d_in[4];
    const float* W1r = (const float*)d_in[5];
    const float* W2l = (const float*)d_in[6];
    const float* b2l = (const float*)d_in[7];
    const float* W2r = (const float*)d_in[8];
    const float* Wi  = (const float*)d_in[9];
    const float* bi  = (const float*)d_in[10];
    /* Wh = d_in[11] unused: hidden state is zero -> only bh contributes */
    const float* bh  = (const float*)d_in[12];
    const float* Wo  = (const float*)d_in[13];
    const float* bo  = (const float*)d_in[14];

    const long long N = in_sizes[0] / DIM;
    const long long E = in_sizes[1] / 2;
    const long long Q = in_sizes[2] / 2;

    float* ws  = (float*)d_ws;
    float* agg = ws;                        // N*128   (later reused as `out`)
    float* cnt = agg + (size_t)N * DIM;     // N
    float* h   = cnt + N;                   // N*128   (later reused as `hid`)
    float* h2  = h + (size_t)N * DIM;       // N*128

    const int* src = ei;
    const int* dst = ei + E;

    const int BT = 256;
    const long long nGemmBlk = (N + 15) / 16;
    const long long scThreads = E * 32;
    const long long dcThreads = Q * 32;

    // 1) zero agg + cnt (contiguous)
    {
        long long n = N * DIM + N;
        zero_kernel<<<(unsigned)((n + BT - 1) / BT), BT, 0, stream>>>(agg, n);
    }
    // 2) scatter x, with degree count
    scatter_kernel<<<(unsigned)((scThreads + BT - 1) / BT), BT, 0, stream>>>(
        x, src, dst, agg, cnt, E, 1);
    // 3) h = relu(mean @ W1l^T + b1l + x @ W1r^T)
    gemm_wmma_kernel<<<(unsigned)nGemmBlk, BT, 0, stream>>>(
        agg, cnt, W1l, x, W1r, b1l, nullptr, h, (int)N, 1);
    // 4) zero agg only
    {
        long long n = N * DIM;
        zero_kernel<<<(unsigned)((n + BT - 1) / BT), BT, 0, stream>>>(agg, n);
    }
    // 5) scatter h (same edges -> cnt already valid)
    scatter_kernel<<<(unsigned)((scThreads + BT - 1) / BT), BT, 0, stream>>>(
        h, src, dst, agg, cnt, E, 0);
    // 6) h2 = mean(h) @ W2l^T + b2l + h @ W2r^T
    gemm_wmma_kernel<<<(unsigned)nGemmBlk, BT, 0, stream>>>(
        agg, cnt, W2l, h, W2r, b2l, nullptr, h2, (int)N, 0);
    // 7) hid = tanh(h2 @ Wi^T + bi + bh)   (stored into h)
    gemm_wmma_kernel<<<(unsigned)nGemmBlk, BT, 0, stream>>>(
        h2, nullptr, Wi, nullptr, nullptr, bi, bh, h, (int)N, 2);
    // 8) out = hid @ Wo^T + bo             (stored into agg)
    gemm_wmma_kernel<<<(unsigned)nGemmBlk, BT, 0, stream>>>(
        h, nullptr, Wo, nullptr, nullptr, bo, nullptr, agg, (int)N, 0);
    // 9) decode: sigmoid(cosine(out[a], out[b]))
    decode_kernel<<<(unsigned)((dcThreads + BT - 1) / BT), BT, 0, stream>>>(
        agg, dec, Q, (float*)d_out);
}